// topk_crossEntrophy_83159156785910
// MI455X (gfx1250) — compile-verified
//
#include <hip/hip_runtime.h>
#include <hip/hip_bf16.h>

typedef float v2f __attribute__((ext_vector_type(2)));
typedef float v4f __attribute__((ext_vector_type(4)));
typedef float v8f __attribute__((ext_vector_type(8)));

#define C_DIM 32000
#define B_DIM 8192
#define K_SEL 5734  /* int(0.7 * 8192) */

// ---- online softmax helpers ------------------------------------------------
__device__ __forceinline__ void online_update(float v, float& m, float& s) {
    float d = v - m;
    if (d <= 0.0f) {
        s += __expf(d);                 // common case: 1 exp
    } else {
        s = s * __expf(-d) + 1.0f;      // new max: rescale
        m = v;
    }
}

__device__ __forceinline__ void merge_ms(float& m, float& s, float m2, float s2) {
    float M = fmaxf(m, m2);
    s = s * __expf(m - M) + s2 * __expf(m2 - M);
    m = M;
}

// ---- kernel 1: per-row NLL = logsumexp(row) - row[target] ------------------
__global__ __launch_bounds__(256)
void row_nll_kernel(const float* __restrict__ inp, const int* __restrict__ tgt,
                    float* __restrict__ loss) {
    const int row = blockIdx.x;
    const int tid = threadIdx.x;
    const float* rp = inp + (size_t)row * C_DIM;

    // issue target-logit load early (thread 0 only)
    float xt = 0.0f;
    if (tid == 0) xt = rp[tgt[row]];

    // 4 independent accumulator chains (one per float4 component) for ILP
    float m0 = -__builtin_inff(), m1 = m0, m2 = m0, m3 = m0;
    float s0 = 0.0f, s1 = 0.0f, s2 = 0.0f, s3 = 0.0f;

    const v4f* rp4 = (const v4f*)rp;
    for (int i = tid; i < C_DIM / 4; i += 256) {
        v4f v = __builtin_nontemporal_load(rp4 + i);  // global_load_b128, TH=NT
        online_update(v.x, m0, s0);
        online_update(v.y, m1, s1);
        online_update(v.z, m2, s2);
        online_update(v.w, m3, s3);
    }
    float m = m0, s = s0;
    merge_ms(m, s, m1, s1);
    merge_ms(m, s, m2, s2);
    merge_ms(m, s, m3, s3);

    // wave32 max-reduce
    float wmax = m;
#pragma unroll
    for (int off = 1; off < 32; off <<= 1)
        wmax = fmaxf(wmax, __shfl_xor(wmax, off));
    float sv = s * __expf(m - wmax);  // rescale to common wave max

    // cross-lane sum via V_WMMA_F32_16X16X4_F32:
    // A[m,k] holds lane values at k in {0,2}; B = ones  ->  D[m,n] = v(m)+v(m+16)
    v2f a; a.x = sv;   a.y = 0.0f;
    v2f b; b.x = 1.0f; b.y = 1.0f;
    v8f acc = {};
    acc = __builtin_amdgcn_wmma_f32_16x16x4_f32(
        /*neg_a=*/false, a, /*neg_b=*/false, b,
        /*c_mod=*/(short)0, acc, /*reuse_a=*/false, /*reuse_b=*/false);
    float p = acc[0] + acc[1] + acc[2] + acc[3] + acc[4] + acc[5] + acc[6] + acc[7];
    float wsum = p + __shfl_xor(p, 16);  // full 32-lane sum in every lane

    // cross-wave combine (8 waves)
    __shared__ float smax[8];
    __shared__ float ssum[8];
    const int wid = tid >> 5;
    if ((tid & 31) == 0) { smax[wid] = wmax; ssum[wid] = wsum; }
    __syncthreads();
    if (tid == 0) {
        float M = smax[0];
#pragma unroll
        for (int i = 1; i < 8; ++i) M = fmaxf(M, smax[i]);
        float S = 0.0f;
#pragma unroll
        for (int i = 0; i < 8; ++i) S += ssum[i] * __expf(smax[i] - M);
        loss[row] = logf(S) + M - xt;
    }
}

// ---- kernel 2: exact top-k mean via 32-bit radix select --------------------
__device__ __forceinline__ unsigned f2key(float f) {
    unsigned u = __float_as_uint(f);
    return (u & 0x80000000u) ? ~u : (u | 0x80000000u);  // order-preserving
}
__device__ __forceinline__ float key2f(unsigned k) {
    unsigned u = (k & 0x80000000u) ? (k & 0x7FFFFFFFu) : ~k;
    return __uint_as_float(u);
}

__global__ __launch_bounds__(1024)
void topk_mean_kernel(const float* __restrict__ loss, float* __restrict__ out) {
    __shared__ unsigned keys[B_DIM];      // 32 KB of the 320 KB WGP LDS
    __shared__ unsigned cnt_sh;
    __shared__ float    fsum_sh;
    __shared__ unsigned cgt_sh;

    const int tid = threadIdx.x;
    for (int i = tid; i < B_DIM; i += 1024) keys[i] = f2key(loss[i]);
    __syncthreads();

    // find key of k-th largest: max prefix with count(key >= prefix) >= k
    unsigned prefix = 0;
    for (int b = 31; b >= 0; --b) {
        unsigned tent = prefix | (1u << b);
        if (tid == 0) cnt_sh = 0;
        __syncthreads();
        unsigned c = 0;
        for (int i = tid; i < B_DIM; i += 1024) c += (keys[i] >= tent) ? 1u : 0u;
#pragma unroll
        for (int off = 1; off < 32; off <<= 1) c += __shfl_xor(c, off);
        if ((tid & 31) == 0) atomicAdd(&cnt_sh, c);
        __syncthreads();
        if (cnt_sh >= (unsigned)K_SEL) prefix = tent;
        __syncthreads();
    }

    // sum of strictly-greater values + tie fill at the threshold
    if (tid == 0) { fsum_sh = 0.0f; cgt_sh = 0; }
    __syncthreads();
    float fs = 0.0f;
    unsigned cg = 0;
    for (int i = tid; i < B_DIM; i += 1024) {
        unsigned kk = keys[i];
        if (kk > prefix) { fs += key2f(kk); cg++; }
    }
#pragma unroll
    for (int off = 1; off < 32; off <<= 1) {
        fs += __shfl_xor(fs, off);
        cg += __shfl_xor(cg, off);
    }
    if ((tid & 31) == 0) { atomicAdd(&fsum_sh, fs); atomicAdd(&cgt_sh, cg); }
    __syncthreads();
    if (tid == 0) {
        float total = fsum_sh + (float)(K_SEL - (int)cgt_sh) * key2f(prefix);
        out[0] = total / (float)K_SEL;
    }
}

// ---- launcher ---------------------------------------------------------------
extern "C" void kernel_launch(void* const* d_in, const int* in_sizes, int n_in,
                              void* d_out, int out_size, void* d_ws, size_t ws_size,
                              hipStream_t stream) {
    const float* inp = (const float*)d_in[0];   // [8192, 32000] fp32
    const int*   tgt = (const int*)d_in[1];     // [8192] int
    float* lossbuf = (float*)d_ws;              // 8192 floats scratch
    float* out     = (float*)d_out;             // scalar fp32

    hipLaunchKernelGGL(row_nll_kernel, dim3(B_DIM), dim3(256), 0, stream,
                       inp, tgt, lossbuf);
    hipLaunchKernelGGL(topk_mean_kernel, dim3(1), dim3(1024), 0, stream,
                       lossbuf, out);
    (void)in_sizes; (void)n_in; (void)out_size; (void)ws_size;
}